// Orthogonal_42339787604006
// MI455X (gfx1250) — compile-verified
//
#include <hip/hip_runtime.h>

// Out = X * M^T where M = H_{u0} * H_{u1} * ... * H_{u511},  H_v = I - 2 v v^T,
// built via compact-WY:  M = I - V*Z,  V[:,k] = U[511-k],
//   Z[k,:] = 2*U[511-k,:] - 2*sum_{i<k} G[k][i] * Z[i,:],
//   G[k][i] = dot(U[511-k], U[511-i]).
// Final GEMM (34 GFLOP, memory-bound at 23.3 TB/s) runs on V_WMMA_F32_16X16X4_F32
// to keep full f32 precision.

#define DDIM 512
#define BATCHN 65536
#define ZB 64               // z-recurrence block size
#define NBLK (DDIM / ZB)
#define LDSW 132            // padded LDS k-stride (132 % 64 == 4 -> conflict-free A reads)

typedef float v2f __attribute__((ext_vector_type(2)));
typedef float v8f __attribute__((ext_vector_type(8)));

// ---------------- K1: Gram matrix in reflection order -------------------------
// G[k][i] = dot(U[511-k], U[511-i])  (full 512x512; symmetric, tiny: 268 MFLOP)
__global__ __launch_bounds__(256) void gram_kernel(const float* __restrict__ U,
                                                   float* __restrict__ G) {
    __shared__ float As[16][17];
    __shared__ float Bs[16][17];
    const int tx = threadIdx.x, ty = threadIdx.y;
    const int k0 = blockIdx.y * 16, i0 = blockIdx.x * 16;
    float acc = 0.0f;
    for (int kk = 0; kk < DDIM; kk += 16) {
        As[ty][tx] = U[(DDIM - 1 - (k0 + ty)) * DDIM + kk + tx];
        Bs[ty][tx] = U[(DDIM - 1 - (i0 + ty)) * DDIM + kk + tx];
        __syncthreads();
#pragma unroll
        for (int t = 0; t < 16; ++t) acc += As[ty][t] * Bs[tx][t];
        __syncthreads();
    }
    G[(k0 + ty) * DDIM + (i0 + tx)] = acc;
}

// ---------------- K2: inter-block Z update (grid-parallel GEMM) ---------------
// Z[start..start+ZB, :] = 2*Urev - 2 * G[rows, 0:start] * Z[0:start, :]
__global__ __launch_bounds__(256) void z_update_kernel(const float* __restrict__ U,
                                                       const float* __restrict__ G,
                                                       float* __restrict__ Z, int start) {
    __shared__ float As[16][17];
    __shared__ float Bs[16][17];
    const int tx = threadIdx.x, ty = threadIdx.y;
    const int m0 = start + blockIdx.y * 16;
    const int n0 = blockIdx.x * 16;
    float acc = 0.0f;
    for (int kk = 0; kk < start; kk += 16) {
        As[ty][tx] = G[(m0 + ty) * DDIM + kk + tx];
        Bs[ty][tx] = Z[(kk + ty) * DDIM + n0 + tx];
        __syncthreads();
#pragma unroll
        for (int t = 0; t < 16; ++t) acc += As[ty][t] * Bs[t][tx];
        __syncthreads();
    }
    const int r = m0 + ty, c = n0 + tx;
    Z[r * DDIM + c] = 2.0f * U[(DDIM - 1 - r) * DDIM + c] - 2.0f * acc;
}

// ---------------- K3: intra-block Z recurrence (1 workgroup) ------------------
// Thread c owns column c; the 64 block rows live in VGPRs, so the sequential
// 2016-FMA triangle never touches memory (G diag tile staged in LDS, uniform reads).
__global__ __launch_bounds__(512) void z_diag_kernel(const float* __restrict__ G,
                                                     float* __restrict__ Z, int start) {
    __shared__ float gs[ZB][ZB];
    const int c = threadIdx.x;
    for (int t = c; t < ZB * ZB; t += 512) {
        const int r = t >> 6, i = t & (ZB - 1);
        gs[r][i] = G[(start + r) * DDIM + (start + i)];
    }
    __syncthreads();
    float acc[ZB];
#pragma unroll
    for (int k = 0; k < ZB; ++k) acc[k] = Z[(start + k) * DDIM + c];
#pragma unroll
    for (int i = 0; i < ZB - 1; ++i) {
        const float m2zi = -2.0f * acc[i];
#pragma unroll
        for (int k = i + 1; k < ZB; ++k) acc[k] = fmaf(gs[k][i], m2zi, acc[k]);
    }
#pragma unroll
    for (int k = 0; k < ZB; ++k) Z[(start + k) * DDIM + c] = acc[k];
}

// ---------------- K4: form M = I - V*Z ---------------------------------------
// M[r][c] = (r==c) - sum_i U[511-i][r] * Z[i][c]
__global__ __launch_bounds__(256) void m_form_kernel(const float* __restrict__ U,
                                                     const float* __restrict__ Z,
                                                     float* __restrict__ Mm) {
    __shared__ float As[16][17];  // As[i_local][r_local] (loaded coalesced along r)
    __shared__ float Bs[16][17];
    const int tx = threadIdx.x, ty = threadIdx.y;
    const int r0 = blockIdx.y * 16, c0 = blockIdx.x * 16;
    float acc = 0.0f;
    for (int kk = 0; kk < DDIM; kk += 16) {
        As[ty][tx] = U[(DDIM - 1 - (kk + ty)) * DDIM + r0 + tx];
        Bs[ty][tx] = Z[(kk + ty) * DDIM + c0 + tx];
        __syncthreads();
#pragma unroll
        for (int t = 0; t < 16; ++t) acc += As[t][ty] * Bs[t][tx];
        __syncthreads();
    }
    const int r = r0 + ty, c = c0 + tx;
    Mm[r * DDIM + c] = (r == c ? 1.0f : 0.0f) - acc;
}

// ---------------- K5: Out = X * M^T  (WMMA f32 16x16x4) -----------------------
// Workgroup = 8 waves, tile 32(batch) x 512(all cols): X read once from HBM,
// staged in LDS in 128-wide k-chunks (padded stride -> bank-conflict-free),
// M strips stream from L2. Each wave: 32x64 tile = 2x4 accumulators.
__global__ __launch_bounds__(256) void xmt_wmma_kernel(const float* __restrict__ X,
                                                       const float* __restrict__ Mm,
                                                       float* __restrict__ Out) {
    __shared__ float Xs[32 * LDSW];
    const int tid  = threadIdx.x;
    const int m0   = blockIdx.x * 32;
    const int wave = tid >> 5;
    const int lane = tid & 31;
    const int n0   = wave * 64;
    const int lr    = lane & 15;          // tile row (A) / tile col (B)
    const int kHalf = (lane >> 4) << 1;   // lanes 16-31 hold K+2,K+3

    v8f acc[2][4];
#pragma unroll
    for (int h = 0; h < 2; ++h)
#pragma unroll
        for (int t = 0; t < 4; ++t) acc[h][t] = {};

    for (int kc = 0; kc < DDIM; kc += 128) {
        __syncthreads();
        // stage X[32][kc..kc+128): 1024 float4 / 256 threads, fully coalesced
#pragma unroll
        for (int it = 0; it < 4; ++it) {
            const int idx = tid + 256 * it;
            const int row = idx >> 5;
            const int c4  = (idx & 31) << 2;
            const float4 v = *reinterpret_cast<const float4*>(
                X + (size_t)(m0 + row) * DDIM + kc + c4);
            *reinterpret_cast<float4*>(&Xs[row * LDSW + c4]) = v;
        }
        __syncthreads();

        for (int k4 = 0; k4 < 128; k4 += 4) {
            const int kO = k4 + kHalf;
            v2f a0, a1;
            a0.x = Xs[lr * LDSW + kO];
            a0.y = Xs[lr * LDSW + kO + 1];
            a1.x = Xs[(16 + lr) * LDSW + kO];
            a1.y = Xs[(16 + lr) * LDSW + kO + 1];
#pragma unroll
            for (int t = 0; t < 4; ++t) {
                const float* mp = Mm + (size_t)(n0 + 16 * t + lr) * DDIM + kc + kO;
                v2f b;
                b.x = mp[0];
                b.y = mp[1];
                acc[0][t] = __builtin_amdgcn_wmma_f32_16x16x4_f32(
                    false, a0, false, b, (short)0, acc[0][t], false, false);
                acc[1][t] = __builtin_amdgcn_wmma_f32_16x16x4_f32(
                    false, a1, false, b, (short)0, acc[1][t], false, false);
            }
        }
    }

    // C/D layout: VGPR j -> row j (lanes 0-15) / row 8+j (lanes 16-31), col = lane&15
    const int rowAdd = (lane < 16) ? 0 : 8;
#pragma unroll
    for (int h = 0; h < 2; ++h)
#pragma unroll
        for (int t = 0; t < 4; ++t)
#pragma unroll
            for (int j = 0; j < 8; ++j) {
                const int r = m0 + 16 * h + j + rowAdd;
                const int c = n0 + 16 * t + lr;
                __builtin_nontemporal_store(acc[h][t][j], Out + (size_t)r * DDIM + c);
            }
}

extern "C" void kernel_launch(void* const* d_in, const int* in_sizes, int n_in,
                              void* d_out, int out_size, void* d_ws, size_t ws_size,
                              hipStream_t stream) {
    (void)in_sizes; (void)n_in; (void)out_size; (void)ws_size;
    const float* X = (const float*)d_in[0];
    const float* U = (const float*)d_in[1];
    float* Out = (float*)d_out;

    float* G  = (float*)d_ws;        // 1 MB
    float* Z  = G + DDIM * DDIM;     // 1 MB
    float* Mm = G;                   // alias: G is dead once Z is complete

    gram_kernel<<<dim3(32, 32), dim3(16, 16), 0, stream>>>(U, G);
    for (int b = 0; b < NBLK; ++b) {
        z_update_kernel<<<dim3(DDIM / 16, ZB / 16), dim3(16, 16), 0, stream>>>(U, G, Z, b * ZB);
        z_diag_kernel<<<1, 512, 0, stream>>>(G, Z, b * ZB);
    }
    m_form_kernel<<<dim3(32, 32), dim3(16, 16), 0, stream>>>(U, Z, Mm);
    xmt_wmma_kernel<<<BATCHN / 32, 256, 0, stream>>>(X, Mm, Out);
}